// ConvCapsuleLayer_87187836109106
// MI455X (gfx1250) — compile-verified
//
#include <hip/hip_runtime.h>
#include <hip/hip_bf16.h>

typedef __attribute__((ext_vector_type(2))) float v2f;
typedef __attribute__((ext_vector_type(8))) float v8f;

#define IC   8
#define ID   16
#define OC   16
#define OD   16
#define KK   3
#define GG   72        // K*K*IC
#define OPG  256       // OC*OD
#define BB   16
#define OHW  49        // 7*7
#define ROWS 784       // B*OHW
#define XCH  136       // IC*(1+ID)
#define HWIN 16

// ---------------------------------------------------------------------------
// a_g[b][g][hw] = acts paired per reference's (IC,K,K) decomposition of g
__global__ void prep_act_kernel(const float* __restrict__ xin, float* __restrict__ ag) {
    int t = blockIdx.x * blockDim.x + threadIdx.x;
    if (t >= BB * GG * OHW) return;
    int hw = t % OHW;
    int g  = (t / OHW) % GG;
    int b  = t / (OHW * GG);
    int y = hw / 7, x = hw % 7;
    int ic2 = g / 9, kh2 = (g % 9) / 3, kw2 = g % 3;
    // acts channel = ic2*17, spatial (kh2+2y, kw2+2x)
    ag[t] = xin[((size_t)b * XCH + (size_t)(ic2 * 17)) * (HWIN * HWIN)
                + (size_t)(kh2 + 2 * y) * HWIN + (kw2 + 2 * x)];
}

__global__ void init_R_kernel(float* __restrict__ R, int n) {
    int t = blockIdx.x * blockDim.x + threadIdx.x;
    if (t < n) R[t] = 1.0f / (float)OC;
}

// ---------------------------------------------------------------------------
// votes[g][row=b*49+hw][o=oc*16+od] = sum_d p[b,g,d,hw]*W[g,o,d] + bias[g,o]
// One wave per 16x16 tile, K=16 via four chained V_WMMA_F32_16X16X4_F32.
__global__ void votes_wmma_kernel(const float* __restrict__ xin,
                                  const float* __restrict__ W,
                                  const float* __restrict__ bias,
                                  float* __restrict__ votes) {
    int wid  = (blockIdx.x * blockDim.x + threadIdx.x) >> 5;
    int lane = threadIdx.x & 31;
    // tiles: g(72) x mt(49) x nt(16)
    int nt = wid & 15;
    int mt = (wid >> 4) % 49;
    int g  = wid / (16 * 49);
    if (g >= GG) return;
    int ic = g & 7, khkw = g >> 3, kh = khkw / 3, kw = khkw % 3;
    int half = lane >> 4;
    int mr   = lane & 15;       // A row within tile (both halves cover m=0..15)
    int nc   = lane & 15;       // B/C/D column within tile
    int n_global = nt * 16 + nc;

    // C init with bias broadcast down rows
    float bb = bias[(size_t)g * OPG + n_global];
    v8f c;
#pragma unroll
    for (int j = 0; j < 8; ++j) c[j] = bb;

    // A source address: p[b,g,d,y,x] = x[b, ic*17+1+d, kh+2y, kw+2x]
    int row  = mt * 16 + mr;            // 0..783
    int bidx = row / OHW, hw = row % OHW;
    int oy = hw / 7, ox = hw % 7;
    int ih = kh + 2 * oy, iw = kw + 2 * ox;
    const float* xbase = xin + ((size_t)bidx * XCH + (size_t)(ic * 17 + 1)) * (HWIN * HWIN)
                             + (size_t)ih * HWIN + iw;          // stride 256 per d
    const float* wbase = W + (size_t)(g * OPG + n_global) * ID; // contiguous in d

#pragma unroll
    for (int kk = 0; kk < 4; ++kk) {
        int k0 = kk * 4 + 2 * half;
        v2f a, bf;
        a[0]  = xbase[(size_t)(k0 + 0) * (HWIN * HWIN)];
        a[1]  = xbase[(size_t)(k0 + 1) * (HWIN * HWIN)];
        bf[0] = wbase[k0 + 0];
        bf[1] = wbase[k0 + 1];
        c = __builtin_amdgcn_wmma_f32_16x16x4_f32(false, a, false, bf,
                                                  (short)0, c, false, false);
    }

    float* vout = votes + ((size_t)g * ROWS + (size_t)mt * 16) * OPG + nt * 16 + nc;
#pragma unroll
    for (int j = 0; j < 8; ++j) {
        int rloc = j + 8 * half;        // D layout: VGPR j -> M = j (+8 for hi half)
        vout[(size_t)rloc * OPG] = c[j];
    }
}

// ---------------------------------------------------------------------------
// Moments: one wave per (b,oc,hw). lane = od(16) x g-half(2).
// Two-pass mean/variance over g (matches reference numerics), then cost,
// a_hat, sigma_prod.
__global__ void moments_kernel(const float* __restrict__ votes,
                               const float* __restrict__ R,
                               const float* __restrict__ ag,
                               const float* __restrict__ beta_v,
                               const float* __restrict__ beta_a,
                               float* __restrict__ u_h,
                               float* __restrict__ sigma,
                               float* __restrict__ a_hat,
                               float* __restrict__ rsp) {
    int wid  = (blockIdx.x * blockDim.x + threadIdx.x) >> 5;
    int lane = threadIdx.x & 31;
    if (wid >= BB * OC * OHW) return;
    int hw = wid % OHW;
    int oc = (wid / OHW) % OC;
    int b  = wid / (OHW * OC);
    int od = lane & 15;
    int gh = lane >> 4;
    int row = b * OHW + hw;

    // pass 1: weighted sum + sum of weights
    float s1 = 0.f, srl = 0.f;
    for (int g = gh * 36; g < gh * 36 + 36; ++g) {
        float rh = R[(((size_t)b * GG + g) * OC + oc) * OHW + hw]
                 * ag[((size_t)b * GG + g) * OHW + hw];
        float v = votes[((size_t)g * ROWS + row) * OPG + oc * 16 + od];
        s1 += rh * v;
        srl += rh;
    }
    s1  += __shfl_xor(s1, 16);
    srl += __shfl_xor(srl, 16);
    float sr = srl;
    float uh = s1 / sr;

    // pass 2: weighted variance about uh
    float s2 = 0.f;
    for (int g = gh * 36; g < gh * 36 + 36; ++g) {
        float rh = R[(((size_t)b * GG + g) * OC + oc) * OHW + hw]
                 * ag[((size_t)b * GG + g) * OHW + hw];
        float v = votes[((size_t)g * ROWS + row) * OPG + oc * 16 + od];
        float d = v - uh;
        s2 += rh * d * d;
    }
    s2 += __shfl_xor(s2, 16);
    float sg = s2 / sr;

    // cost over od, sigma product over od (butterfly within 16-lane half)
    float cost = (beta_v[oc * OHW + hw] + 0.5f * __logf(sg)) * sr;
    float csum = cost;
    float prod = 2.0f * 3.1416f * sg;
#pragma unroll
    for (int off = 8; off >= 1; off >>= 1) {
        csum += __shfl_xor(csum, off);
        prod *= __shfl_xor(prod, off);
    }
    float ah = 1.0f / (1.0f + __expf(-0.01f * (beta_a[oc * OHW + hw] - csum)));

    size_t uix = (((size_t)b * OC + oc) * OD + od) * OHW + hw;
    if (lane < 16) {
        u_h[uix]   = uh;
        sigma[uix] = sg;
    }
    if (lane == 0) {
        size_t aix = ((size_t)b * OC + oc) * OHW + hw;
        a_hat[aix] = ah;
        rsp[aix]   = rsqrtf(prod);
    }
}

// ---------------------------------------------------------------------------
// num[b][g][oc][hw] = a_hat * exp(-(sum_od diff^2/(2 sigma)) * rsqrt(sigma_prod))
__global__ void pc_kernel(const float* __restrict__ votes,
                          const float* __restrict__ u_h,
                          const float* __restrict__ sigma,
                          const float* __restrict__ a_hat,
                          const float* __restrict__ rsp,
                          float* __restrict__ num) {
    size_t t = (size_t)blockIdx.x * blockDim.x + threadIdx.x;
    if (t >= (size_t)BB * GG * OC * OHW) return;
    int hw = t % OHW;
    int oc = (t / OHW) % OC;
    int g  = (t / (OHW * OC)) % GG;
    int b  = t / ((size_t)OHW * OC * GG);
    int row = b * OHW + hw;
    const float* v = votes + ((size_t)g * ROWS + row) * OPG + oc * 16;
    float s = 0.f;
#pragma unroll
    for (int od = 0; od < OD; ++od) {
        size_t uix = (((size_t)b * OC + oc) * OD + od) * OHW + hw;
        float d = v[od] - u_h[uix];
        s += 0.5f * d * d / sigma[uix];
    }
    size_t aix = ((size_t)b * OC + oc) * OHW + hw;
    num[t] = a_hat[aix] * __expf(-s * rsp[aix]);
}

// Deterministic per-(b,g) reduction of num over (oc,hw): 1152 blocks
__global__ void denom_kernel(const float* __restrict__ num, float* __restrict__ denom) {
    __shared__ float sdata[256];
    int bg = blockIdx.x;
    float s = 0.f;
    for (int i = threadIdx.x; i < OC * OHW; i += 256)
        s += num[(size_t)bg * (OC * OHW) + i];
    sdata[threadIdx.x] = s;
    __syncthreads();
    for (int off = 128; off > 0; off >>= 1) {
        if ((int)threadIdx.x < off) sdata[threadIdx.x] += sdata[threadIdx.x + off];
        __syncthreads();
    }
    if (threadIdx.x == 0) denom[bg] = sdata[0];
}

__global__ void update_R_kernel(const float* __restrict__ num,
                                const float* __restrict__ denom,
                                float* __restrict__ R) {
    size_t t = (size_t)blockIdx.x * blockDim.x + threadIdx.x;
    if (t >= (size_t)BB * GG * OC * OHW) return;
    int bg = (int)(t / (OHW * OC));   // (b*72+g)
    R[t] = num[t] / denom[bg];
}

// ---------------------------------------------------------------------------
// out[b][oc*17 + c][hw]: c=0 -> a_hat ; c=1..16 -> squash(u_h)
__global__ void output_kernel(const float* __restrict__ u_h,
                              const float* __restrict__ a_hat,
                              float* __restrict__ out) {
    int t = blockIdx.x * blockDim.x + threadIdx.x;
    if (t >= BB * OC * OHW) return;
    int hw = t % OHW;
    int oc = (t / OHW) % OC;
    int b  = t / (OHW * OC);
    float u[OD];
    float sn = 0.f;
#pragma unroll
    for (int od = 0; od < OD; ++od) {
        u[od] = u_h[(((size_t)b * OC + oc) * OD + od) * OHW + hw];
        sn += u[od] * u[od];
    }
    float scale = sn / (1.0f + sn) * rsqrtf(sn);
    size_t obase = ((size_t)b * (OC * 17) + (size_t)oc * 17) * OHW + hw;
    out[obase] = a_hat[((size_t)b * OC + oc) * OHW + hw];
#pragma unroll
    for (int od = 0; od < OD; ++od)
        out[obase + (size_t)(1 + od) * OHW] = u[od] * scale;
}

// ---------------------------------------------------------------------------
extern "C" void kernel_launch(void* const* d_in, const int* in_sizes, int n_in,
                              void* d_out, int out_size, void* d_ws, size_t ws_size,
                              hipStream_t stream) {
    const float* xin    = (const float*)d_in[0];
    const float* W      = (const float*)d_in[1];
    const float* bias   = (const float*)d_in[2];
    const float* beta_v = (const float*)d_in[3];
    const float* beta_a = (const float*)d_in[4];
    float* out = (float*)d_out;

    float* ws = (float*)d_ws;
    size_t off = 0;
    float* votes = ws + off; off += (size_t)GG * ROWS * OPG;      // 14,450,688
    float* R     = ws + off; off += (size_t)BB * GG * OC * OHW;   //    903,168
    float* num   = ws + off; off += (size_t)BB * GG * OC * OHW;   //    903,168
    float* ag    = ws + off; off += (size_t)BB * GG * OHW;        //     56,448
    float* u_h   = ws + off; off += (size_t)BB * OC * OD * OHW;   //    200,704
    float* sigma = ws + off; off += (size_t)BB * OC * OD * OHW;   //    200,704
    float* a_hat = ws + off; off += (size_t)BB * OC * OHW;        //     12,544
    float* rsp   = ws + off; off += (size_t)BB * OC * OHW;        //     12,544
    float* denom = ws + off; off += (size_t)BB * GG;              //      1,152

    const int TB = 256;

    // activations + R init
    {
        int n = BB * GG * OHW;
        prep_act_kernel<<<(n + TB - 1) / TB, TB, 0, stream>>>(xin, ag);
    }
    {
        int n = BB * GG * OC * OHW;
        init_R_kernel<<<(n + TB - 1) / TB, TB, 0, stream>>>(R, n);
    }
    // votes GEMM: 72*49*16 wave-tiles, 8 waves per block
    {
        int waves = GG * 49 * 16;
        votes_wmma_kernel<<<(waves * 32 + TB - 1) / TB, TB, 0, stream>>>(xin, W, bias, votes);
    }
    // EM routing
    int npos  = BB * OC * OHW;                 // 12,544 waves for moments
    int nnum  = BB * GG * OC * OHW;            // 903,168
    for (int it = 0; it < 3; ++it) {
        moments_kernel<<<(npos * 32 + TB - 1) / TB, TB, 0, stream>>>(
            votes, R, ag, beta_v, beta_a, u_h, sigma, a_hat, rsp);
        if (it < 2) {
            pc_kernel<<<(nnum + TB - 1) / TB, TB, 0, stream>>>(
                votes, u_h, sigma, a_hat, rsp, num);
            denom_kernel<<<BB * GG, TB, 0, stream>>>(num, denom);
            update_R_kernel<<<(nnum + TB - 1) / TB, TB, 0, stream>>>(num, denom, R);
        }
    }
    // final output
    output_kernel<<<(npos + TB - 1) / TB, TB, 0, stream>>>(u_h, a_hat, out);

    (void)in_sizes; (void)n_in; (void)out_size; (void)ws_size;
}